// DGCNN2_16939351016241
// MI455X (gfx1250) — compile-verified
//
#include <hip/hip_runtime.h>
#include <hip/hip_bf16.h>

typedef float v2f __attribute__((ext_vector_type(2)));
typedef float v8f __attribute__((ext_vector_type(8)));

// Problem constants
#define Bb   8
#define Nn   2000
#define Cc   128
#define Kk   10
#define C2c  256
#define OUTC 128
#define SP   (Nn*Kk)          // 20000 spatial per (b,c)
#define ECNT (Bb*C2c*SP)      // 40,960,000 elements in e/h buffers

// ---------------------------------------------------------------------------
// 1. row squared norms xx[b*N+n] = sum_c x^2
// ---------------------------------------------------------------------------
__global__ void k_rownorm(const float* __restrict__ x, float* __restrict__ xx) {
    int i = blockIdx.x * blockDim.x + threadIdx.x;
    if (i >= Bb * Nn) return;
    const float* p = x + (size_t)i * Cc;
    float s = 0.f;
    #pragma unroll 8
    for (int c = 0; c < Cc; ++c) { float v = p[c]; s += v * v; }
    xx[i] = s;
}

// ---------------------------------------------------------------------------
// 2. neg_dist[b][m][n] = 2*<x_m,x_n> - xx_m - xx_n  via fp32 WMMA 16x16x4
//    5x5 register blocking: one wave -> 80x80 output (25 accumulators),
//    10 float2 loads feed 25 WMMAs per K-step.
//    8 batches * 25*25 wave-jobs = 5000 waves = 1250 blocks * 4 waves.
// ---------------------------------------------------------------------------
__global__ __launch_bounds__(128) void k_negdist(const float* __restrict__ x,
                                                 const float* __restrict__ xx,
                                                 float* __restrict__ nd) {
    const int wave = threadIdx.x >> 5;
    const int lane = threadIdx.x & 31;
    const int lo   = lane & 15;
    const bool hi  = lane >= 16;
    int j  = blockIdx.x * 4 + wave;         // job id, exact
    int gn = j % 25;
    int gm = (j / 25) % 25;
    int b  = j / 625;

    const float* X = x + (size_t)b * Nn * Cc;
    const float* ap[5];
    const float* bp[5];
    #pragma unroll
    for (int i = 0; i < 5; ++i) {
        ap[i] = X + (size_t)((gm * 5 + i) * 16 + lo) * Cc + (hi ? 2 : 0);
        bp[i] = X + (size_t)((gn * 5 + i) * 16 + lo) * Cc + (hi ? 2 : 0);
    }

    v8f acc[5][5] = {};
    #pragma unroll 2
    for (int kc = 0; kc < Cc; kc += 4) {
        v2f af[5];
        #pragma unroll
        for (int i = 0; i < 5; ++i) af[i] = *(const v2f*)(ap[i] + kc);
        #pragma unroll
        for (int jj = 0; jj < 5; ++jj) {
            v2f bv = *(const v2f*)(bp[jj] + kc);
            #pragma unroll
            for (int i = 0; i < 5; ++i)
                acc[i][jj] = __builtin_amdgcn_wmma_f32_16x16x4_f32(
                    false, af[i], false, bv, (short)0, acc[i][jj], false, false);
        }
    }

    const float* xxb = xx + b * Nn;
    float* out = nd + (size_t)b * Nn * Nn;
    #pragma unroll
    for (int i = 0; i < 5; ++i) {
        #pragma unroll
        for (int jj = 0; jj < 5; ++jj) {
            int nc0 = (gn * 5 + jj) * 16 + lo;
            float xn = xxb[nc0];
            #pragma unroll
            for (int r = 0; r < 8; ++r) {
                int m = (gm * 5 + i) * 16 + r + (hi ? 8 : 0);
                out[(size_t)m * Nn + nc0] = 2.0f * acc[i][jj][r] - xxb[m] - xn;
            }
        }
    }
}

// ---------------------------------------------------------------------------
// 3. top-K (K=10) per row; insertion sort keeps jax.lax.top_k tie-breaking
// ---------------------------------------------------------------------------
__global__ void k_topk(const float* __restrict__ nd, int* __restrict__ idx) {
    int i = blockIdx.x * blockDim.x + threadIdx.x;
    if (i >= Bb * Nn) return;
    const float* row = nd + (size_t)i * Nn;
    float val[Kk]; int ind[Kk];
    #pragma unroll
    for (int j = 0; j < Kk; ++j) { val[j] = -3.4e38f; ind[j] = 0; }
    for (int m = 0; m < Nn; ++m) {
        float cv = row[m]; int ci = m;
        #pragma unroll
        for (int j = 0; j < Kk; ++j) {
            if (cv > val[j]) {
                float tv = val[j]; val[j] = cv; cv = tv;
                int   ti = ind[j]; ind[j] = ci; ci = ti;
            }
        }
    }
    int* o = idx + (size_t)i * Kk;
    #pragma unroll
    for (int j = 0; j < Kk; ++j) o[j] = ind[j];
}

// ---------------------------------------------------------------------------
// 4. build e (B,C2,N,K): [0..C) = x[idx]-x[n], [C..2C) = x[n]
// ---------------------------------------------------------------------------
__global__ void k_builde(const float* __restrict__ x, const int* __restrict__ idx,
                         float* __restrict__ E) {
    int i = blockIdx.x * blockDim.x + threadIdx.x;
    if (i >= ECNT) return;
    int k = i % Kk;
    int n = (i / Kk) % Nn;
    int c = (i / SP) % C2c;
    int b = i / (SP * C2c);
    const float* xb = x + (size_t)b * Nn * Cc;
    float v;
    if (c < Cc) {
        int j = idx[(size_t)(b * Nn + n) * Kk + k];
        v = xb[(size_t)j * Cc + c] - xb[(size_t)n * Cc + c];
    } else {
        v = xb[(size_t)n * Cc + (c - Cc)];
    }
    E[i] = v;
}

// ---------------------------------------------------------------------------
// 5. per-(b,c) mean/var over the 20000 spatial elements (instance norm stats)
// ---------------------------------------------------------------------------
__global__ __launch_bounds__(256) void k_mvreduce(const float* __restrict__ src,
                                                  float2* __restrict__ mv) {
    int bc = blockIdx.x;                       // 2048 blocks
    const float* p = src + (size_t)bc * SP;
    float s = 0.f, q = 0.f;
    for (int i = threadIdx.x; i < SP; i += 256) { float v = p[i]; s += v; q += v * v; }
    __shared__ float s1[256], s2[256];
    s1[threadIdx.x] = s; s2[threadIdx.x] = q; __syncthreads();
    for (int st = 128; st > 0; st >>= 1) {
        if (threadIdx.x < st) { s1[threadIdx.x] += s1[threadIdx.x + st];
                                s2[threadIdx.x] += s2[threadIdx.x + st]; }
        __syncthreads();
    }
    if (threadIdx.x == 0) {
        float m = s1[0] / (float)SP;
        float v = s2[0] / (float)SP - m * m;
        mv[bc] = make_float2(m, v < 0.f ? 0.f : v);
    }
}

// batchnorm stats after instance norm: mean==0, var = mean_b(v/(v+1e-3))
__global__ void k_bncombine(const float2* __restrict__ mv, float* __restrict__ bns) {
    int c = threadIdx.x;
    if (c >= C2c) return;
    float a = 0.f;
    #pragma unroll
    for (int b = 0; b < Bb; ++b) { float v = mv[b * C2c + c].y; a += v / (v + 1e-3f); }
    bns[c] = rsqrtf(a / (float)Bb + 1e-5f);
}

// fused inorm + bnorm + relu
__global__ void k_normapply(const float* __restrict__ src, float* __restrict__ dst,
                            const float2* __restrict__ mv, const float* __restrict__ bns,
                            const float* __restrict__ g, const float* __restrict__ beta) {
    int i = blockIdx.x * blockDim.x + threadIdx.x;
    if (i >= ECNT) return;
    int bc = i / SP;
    int c  = bc % C2c;
    float2 m = mv[bc];
    float y = (src[i] - m.x) * rsqrtf(m.y + 1e-3f) * bns[c] * g[c] + beta[c];
    dst[i] = y > 0.f ? y : 0.f;
}

// ---------------------------------------------------------------------------
// 6. conv1x1: out[b][o][s] = sum_c w[o][c] * in[b][c][s] + bias[o] (+ add)
//    4(M)x5(S) register blocking: 4 weight frags + 5 column pairs feed
//    20 WMMAs per K-step. 8*4*250 = 8000 waves = 2000 blocks.
// ---------------------------------------------------------------------------
__global__ __launch_bounds__(128) void k_conv1x1(const float* __restrict__ in,
                                                 const float* __restrict__ w,
                                                 const float* __restrict__ bias,
                                                 const float* __restrict__ addp,
                                                 float* __restrict__ out) {
    const int wave = threadIdx.x >> 5;
    const int lane = threadIdx.x & 31;
    const int lo   = lane & 15;
    const bool hi  = lane >= 16;
    int j  = blockIdx.x * 4 + wave;          // job id, exact
    int sg = j % 250;
    int mg = (j / 250) % 4;
    int b  = j / 1000;

    const float* ap[4];
    #pragma unroll
    for (int i = 0; i < 4; ++i)
        ap[i] = w + (size_t)((mg * 4 + i) * 16 + lo) * C2c + (hi ? 2 : 0);
    const float* bcol[5];
    #pragma unroll
    for (int jj = 0; jj < 5; ++jj)
        bcol[jj] = in + (size_t)b * C2c * SP + (sg * 5 + jj) * 16 + lo;

    v8f acc[4][5] = {};
    #pragma unroll 2
    for (int kc = 0; kc < C2c; kc += 4) {
        int kb = kc + (hi ? 2 : 0);
        v2f af[4];
        #pragma unroll
        for (int i = 0; i < 4; ++i) af[i] = *(const v2f*)(ap[i] + kc);
        #pragma unroll
        for (int jj = 0; jj < 5; ++jj) {
            v2f bv;
            bv.x = bcol[jj][(size_t)kb * SP];
            bv.y = bcol[jj][(size_t)(kb + 1) * SP];
            if (kc + 16 < C2c)
                __builtin_prefetch(bcol[jj] + (size_t)(kb + 16) * SP, 0, 1);
            #pragma unroll
            for (int i = 0; i < 4; ++i)
                acc[i][jj] = __builtin_amdgcn_wmma_f32_16x16x4_f32(
                    false, af[i], false, bv, (short)0, acc[i][jj], false, false);
        }
    }
    #pragma unroll
    for (int i = 0; i < 4; ++i) {
        #pragma unroll
        for (int r = 0; r < 8; ++r) {
            int o = (mg * 4 + i) * 16 + r + (hi ? 8 : 0);
            float bias_v = bias[o];
            #pragma unroll
            for (int jj = 0; jj < 5; ++jj) {
                size_t oi = (size_t)(b * C2c + o) * SP + (sg * 5 + jj) * 16 + lo;
                float v = acc[i][jj][r] + bias_v;
                if (addp) v += addp[oi];
                out[oi] = v;
            }
        }
    }
}

// ---------------------------------------------------------------------------
// 7. neighbor gate: ncomp (B,2,C2,N) = {max_k, mean_k} of e
// ---------------------------------------------------------------------------
__global__ void k_ncomp(const float* __restrict__ E, float* __restrict__ nc) {
    int i = blockIdx.x * blockDim.x + threadIdx.x;     // 4,096,000 exact
    int n = i % Nn;
    int c = (i / Nn) % C2c;
    int b = i / (Nn * C2c);
    const float* p = E + ((size_t)(b * C2c + c) * Nn + n) * Kk;
    float mx = -3.4e38f, s = 0.f;
    #pragma unroll
    for (int k = 0; k < Kk; ++k) { float v = p[k]; mx = v > mx ? v : mx; s += v; }
    nc[((size_t)(b * 2 + 0) * C2c + c) * Nn + n] = mx;
    nc[((size_t)(b * 2 + 1) * C2c + c) * Nn + n] = s * (1.0f / Kk);
}

// 3x3 conv over (C2,N) spatial, 2 input channels, SAME pad + partial stats
__global__ __launch_bounds__(256) void k_conv3n(const float* __restrict__ nc,
                                                const float* __restrict__ wgt,
                                                float* __restrict__ raw,
                                                float2* __restrict__ part) {
    int i = blockIdx.x * blockDim.x + threadIdx.x;     // 16000 blocks exact
    int n = i % Nn;
    int c = (i / Nn) % C2c;
    int b = i / (Nn * C2c);
    float acc = 0.f;
    #pragma unroll
    for (int ch = 0; ch < 2; ++ch)
        #pragma unroll
        for (int dy = -1; dy <= 1; ++dy)
            #pragma unroll
            for (int dx = -1; dx <= 1; ++dx) {
                int cc = c + dy, nn2 = n + dx;
                if (cc >= 0 && cc < C2c && nn2 >= 0 && nn2 < Nn)
                    acc += wgt[ch * 9 + (dy + 1) * 3 + (dx + 1)] *
                           nc[((size_t)(b * 2 + ch) * C2c + cc) * Nn + nn2];
            }
    raw[(size_t)(b * C2c + c) * Nn + n] = acc;
    __shared__ float s1[256], s2[256];
    s1[threadIdx.x] = acc; s2[threadIdx.x] = acc * acc; __syncthreads();
    for (int st = 128; st > 0; st >>= 1) {
        if (threadIdx.x < st) { s1[threadIdx.x] += s1[threadIdx.x + st];
                                s2[threadIdx.x] += s2[threadIdx.x + st]; }
        __syncthreads();
    }
    if (threadIdx.x == 0) part[blockIdx.x] = make_float2(s1[0], s2[0]);
}

// deterministic final reduce of partials -> affine (A,B) for sigmoid stage
__global__ __launch_bounds__(256) void k_finstat(const float2* __restrict__ part,
                                                 int nparts, float cnt,
                                                 const float* __restrict__ g,
                                                 const float* __restrict__ beta,
                                                 float* __restrict__ gst) {
    float s = 0.f, q = 0.f;
    for (int i = threadIdx.x; i < nparts; i += 256) { float2 p = part[i]; s += p.x; q += p.y; }
    __shared__ float s1[256], s2[256];
    s1[threadIdx.x] = s; s2[threadIdx.x] = q; __syncthreads();
    for (int st = 128; st > 0; st >>= 1) {
        if (threadIdx.x < st) { s1[threadIdx.x] += s1[threadIdx.x + st];
                                s2[threadIdx.x] += s2[threadIdx.x + st]; }
        __syncthreads();
    }
    if (threadIdx.x == 0) {
        float m = s1[0] / cnt;
        float v = s2[0] / cnt - m * m; if (v < 0.f) v = 0.f;
        float A = g[0] * rsqrtf(v + 1e-5f);
        gst[0] = A;
        gst[1] = beta[0] - m * A;
    }
}

__global__ void k_sigmoid(float* __restrict__ x, int n, const float* __restrict__ gst) {
    int i = blockIdx.x * blockDim.x + threadIdx.x;
    if (i >= n) return;
    float y = x[i] * gst[0] + gst[1];
    x[i] = 1.0f / (1.0f + expf(-y));
}

// ---------------------------------------------------------------------------
// 8. spatial gate: scomp (B,2,N,K) = {max_c, mean_c} of e
// ---------------------------------------------------------------------------
__global__ void k_scomp(const float* __restrict__ E, float* __restrict__ sc) {
    int i = blockIdx.x * blockDim.x + threadIdx.x;     // 160000 exact
    int k = i % Kk;
    int n = (i / Kk) % Nn;
    int b = i / (Nn * Kk);
    float mx = -3.4e38f, s = 0.f;
    for (int c = 0; c < C2c; ++c) {
        float v = E[((size_t)(b * C2c + c) * Nn + n) * Kk + k];
        mx = v > mx ? v : mx; s += v;
    }
    sc[((size_t)(b * 2 + 0) * Nn + n) * Kk + k] = mx;
    sc[((size_t)(b * 2 + 1) * Nn + n) * Kk + k] = s * (1.0f / C2c);
}

__global__ __launch_bounds__(256) void k_conv3s(const float* __restrict__ sc,
                                                const float* __restrict__ wgt,
                                                float* __restrict__ raw,
                                                float2* __restrict__ part) {
    int i = blockIdx.x * blockDim.x + threadIdx.x;     // 625 blocks exact
    int k = i % Kk;
    int n = (i / Kk) % Nn;
    int b = i / (Nn * Kk);
    float acc = 0.f;
    #pragma unroll
    for (int ch = 0; ch < 2; ++ch)
        #pragma unroll
        for (int dy = -1; dy <= 1; ++dy)
            #pragma unroll
            for (int dx = -1; dx <= 1; ++dx) {
                int nn2 = n + dy, kk2 = k + dx;
                if (nn2 >= 0 && nn2 < Nn && kk2 >= 0 && kk2 < Kk)
                    acc += wgt[ch * 9 + (dy + 1) * 3 + (dx + 1)] *
                           sc[((size_t)(b * 2 + ch) * Nn + nn2) * Kk + kk2];
            }
    raw[(size_t)(b * Nn + n) * Kk + k] = acc;
    __shared__ float s1[256], s2[256];
    s1[threadIdx.x] = acc; s2[threadIdx.x] = acc * acc; __syncthreads();
    for (int st = 128; st > 0; st >>= 1) {
        if (threadIdx.x < st) { s1[threadIdx.x] += s1[threadIdx.x + st];
                                s2[threadIdx.x] += s2[threadIdx.x + st]; }
        __syncthreads();
    }
    if (threadIdx.x == 0) part[blockIdx.x] = make_float2(s1[0], s2[0]);
}

// ---------------------------------------------------------------------------
// 9. fused:  max_k[ x1*(w0*e*ns + w1*e*ss) + (w0+w1)*e ]  -> (b*N+n, c) layout
// ---------------------------------------------------------------------------
__global__ void k_fusedmax(const float* __restrict__ E, const float* __restrict__ X1,
                           const float* __restrict__ ns, const float* __restrict__ ss,
                           const float* __restrict__ wmix, float* __restrict__ outt) {
    int i = blockIdx.x * blockDim.x + threadIdx.x;     // 4,096,000 exact
    int n = i % Nn;
    int c = (i / Nn) % C2c;
    int b = i / (Nn * C2c);
    float w0 = wmix[0], w1 = wmix[1];
    float nsv = ns[(size_t)(b * C2c + c) * Nn + n];
    const float* ep = E  + ((size_t)(b * C2c + c) * Nn + n) * Kk;
    const float* xp = X1 + ((size_t)(b * C2c + c) * Nn + n) * Kk;
    const float* sp = ss + (size_t)(b * Nn + n) * Kk;
    float mx = -3.4e38f;
    #pragma unroll
    for (int k = 0; k < Kk; ++k) {
        float e = ep[k];
        float t = xp[k] * (w0 * e * nsv + w1 * e * sp[k]) + (w0 + w1) * e;
        mx = t > mx ? t : mx;
    }
    outt[(size_t)(b * Nn + n) * C2c + c] = mx;
}

// ---------------------------------------------------------------------------
// 10. final GEMM: (16000 x 256) @ (256 x 128) + bias via WMMA f32.
//     One wave computes the full 8-tile output row strip: the streaming A
//     matrix is fragment-loaded once per K-step for 8 WMMAs (W stays in L2).
//     1000 waves = 250 blocks.
// ---------------------------------------------------------------------------
__global__ __launch_bounds__(128) void k_fingemm(const float* __restrict__ A,
                                                 const float* __restrict__ W,
                                                 const float* __restrict__ bias,
                                                 float* __restrict__ out) {
    const int wave = threadIdx.x >> 5;
    const int lane = threadIdx.x & 31;
    const int lo   = lane & 15;
    const bool hi  = lane >= 16;
    int tm = blockIdx.x * 4 + wave;          // 250*4 = 1000 row tiles exact

    const float* ap = A + (size_t)(tm * 16 + lo) * C2c + (hi ? 2 : 0);

    v8f acc[8] = {};
    #pragma unroll 4
    for (int kc = 0; kc < C2c; kc += 4) {
        v2f a = *(const v2f*)(ap + kc);
        int kb = kc + (hi ? 2 : 0);
        #pragma unroll
        for (int tn = 0; tn < 8; ++tn) {
            int col = tn * 16 + lo;
            v2f bv;
            bv.x = W[(size_t)kb * OUTC + col];
            bv.y = W[(size_t)(kb + 1) * OUTC + col];
            acc[tn] = __builtin_amdgcn_wmma_f32_16x16x4_f32(
                false, a, false, bv, (short)0, acc[tn], false, false);
        }
    }
    #pragma unroll
    for (int tn = 0; tn < 8; ++tn) {
        int col = tn * 16 + lo;
        float bias_v = bias[col];
        #pragma unroll
        for (int r = 0; r < 8; ++r) {
            int m = tm * 16 + r + (hi ? 8 : 0);
            out[(size_t)m * OUTC + col] = acc[tn][r] + bias_v;
        }
    }
}

// ---------------------------------------------------------------------------
extern "C" void kernel_launch(void* const* d_in, const int* in_sizes, int n_in,
                              void* d_out, int out_size, void* d_ws, size_t ws_size,
                              hipStream_t stream) {
    const float* x     = (const float*)d_in[0];
    const float* bn1_g = (const float*)d_in[1];
    const float* bn1_b = (const float*)d_in[2];
    const float* w1    = (const float*)d_in[3];
    const float* b1    = (const float*)d_in[4];
    const float* bn2_g = (const float*)d_in[5];
    const float* bn2_b = (const float*)d_in[6];
    const float* w2    = (const float*)d_in[7];
    const float* b2    = (const float*)d_in[8];
    const float* ng_w  = (const float*)d_in[9];
    const float* ng_g  = (const float*)d_in[10];
    const float* ng_b  = (const float*)d_in[11];
    const float* sg_w  = (const float*)d_in[12];
    const float* sg_g  = (const float*)d_in[13];
    const float* sg_b  = (const float*)d_in[14];
    const float* lin_w = (const float*)d_in[15];
    const float* lin_b = (const float*)d_in[16];
    const float* wmix  = (const float*)d_in[17];

    char* ws = (char*)d_ws;
    const size_t F = sizeof(float);
    size_t ofE   = 0;
    size_t ofH   = ofE  + (size_t)ECNT * F;                 // also negdist alias
    size_t ofC   = ofH  + (size_t)ECNT * F;
    size_t ofXX  = ofC  + (size_t)ECNT * F;
    size_t ofIDX = ofXX + (size_t)Bb * Nn * F;
    size_t ofMV  = ofIDX + (size_t)Bb * Nn * Kk * sizeof(int);
    size_t ofBNS = ofMV + (size_t)Bb * C2c * sizeof(float2);
    // buffers alive only after conv2: alias inside the C region
    size_t ofNC  = ofC;
    size_t ofNSR = ofNC  + (size_t)Bb * 2 * C2c * Nn * F;
    size_t ofSC  = ofNSR + (size_t)Bb * C2c * Nn * F;
    size_t ofSSR = ofSC  + (size_t)Bb * 2 * Nn * Kk * F;
    size_t ofOT  = ofSSR + (size_t)Bb * Nn * Kk * F;
    size_t ofPN  = ofOT  + (size_t)Bb * Nn * C2c * F;
    size_t ofPS  = ofPN  + 16000 * sizeof(float2);
    size_t ofGN  = ofPS  + 625 * sizeof(float2);
    size_t ofGS  = ofGN  + 2 * F;

    float*  E    = (float*)(ws + ofE);
    float*  H    = (float*)(ws + ofH);    // negdist -> h1 -> x1
    float*  CB   = (float*)(ws + ofC);    // conv1 out / h2
    float*  XX   = (float*)(ws + ofXX);
    int*    IDX  = (int*)  (ws + ofIDX);
    float2* MV   = (float2*)(ws + ofMV);
    float*  BNS  = (float*)(ws + ofBNS);
    float*  NC   = (float*)(ws + ofNC);
    float*  NSR  = (float*)(ws + ofNSR);
    float*  SC   = (float*)(ws + ofSC);
    float*  SSR  = (float*)(ws + ofSSR);
    float*  OT   = (float*)(ws + ofOT);
    float2* PN   = (float2*)(ws + ofPN);
    float2* PS   = (float2*)(ws + ofPS);
    float*  GN   = (float*)(ws + ofGN);
    float*  GS   = (float*)(ws + ofGS);
    float*  ND   = H;                     // negdist aliases H region

    // kNN graph
    k_rownorm<<<(Bb * Nn + 255) / 256, 256, 0, stream>>>(x, XX);
    k_negdist<<<1250, 128, 0, stream>>>(x, XX, ND);
    k_topk<<<(Bb * Nn + 255) / 256, 256, 0, stream>>>(ND, IDX);
    k_builde<<<(ECNT + 255) / 256, 256, 0, stream>>>(x, IDX, E);

    // norm1 + relu : E -> H  (overwrites negdist)
    k_mvreduce<<<Bb * C2c, 256, 0, stream>>>(E, MV);
    k_bncombine<<<1, 256, 0, stream>>>(MV, BNS);
    k_normapply<<<(ECNT + 255) / 256, 256, 0, stream>>>(E, H, MV, BNS, bn1_g, bn1_b);

    // conv1 : H -> CB
    k_conv1x1<<<2000, 128, 0, stream>>>(H, w1, b1, nullptr, CB);

    // norm2 + relu : CB -> CB (in place)
    k_mvreduce<<<Bb * C2c, 256, 0, stream>>>(CB, MV);
    k_bncombine<<<1, 256, 0, stream>>>(MV, BNS);
    k_normapply<<<(ECNT + 255) / 256, 256, 0, stream>>>(CB, CB, MV, BNS, bn2_g, bn2_b);

    // conv2 + e : CB -> H   (H now holds x1)
    k_conv1x1<<<2000, 128, 0, stream>>>(CB, w2, b2, E, H);

    // neighbor gate ns  (CB region reused)
    k_ncomp<<<16000, 256, 0, stream>>>(E, NC);
    k_conv3n<<<16000, 256, 0, stream>>>(NC, ng_w, NSR, PN);
    k_finstat<<<1, 256, 0, stream>>>(PN, 16000, (float)(Bb * C2c * Nn), ng_g, ng_b, GN);
    k_sigmoid<<<16000, 256, 0, stream>>>(NSR, Bb * C2c * Nn, GN);

    // spatial gate ss
    k_scomp<<<625, 256, 0, stream>>>(E, SC);
    k_conv3s<<<625, 256, 0, stream>>>(SC, sg_w, SSR, PS);
    k_finstat<<<1, 256, 0, stream>>>(PS, 625, (float)(Bb * Nn * Kk), sg_g, sg_b, GS);
    k_sigmoid<<<625, 256, 0, stream>>>(SSR, Bb * Nn * Kk, GS);

    // fused mix + max over K  -> (b*N+n, c)
    k_fusedmax<<<16000, 256, 0, stream>>>(E, H, NSR, SSR, wmix, OT);

    // final linear -> d_out (B,N,OUT)
    k_fingemm<<<250, 128, 0, stream>>>(OT, lin_w, lin_b, (float*)d_out);

    (void)in_sizes; (void)n_in; (void)out_size; (void)ws_size;
}